// ToeplitzSingleShotInhibition_66468913873358
// MI455X (gfx1250) — compile-verified
//
#include <hip/hip_runtime.h>

// CDNA5 / gfx1250: banded circulant channel "Toeplitz inhibition"
// out[b,d,hw] = sum_c act[b,c,hw] * filt[31 + circ(d-c)], band |d-c| <= 31.
// Per-batch GEMM with V_WMMA_F32_16X16X4_F32 (full fp32), exploiting the
// 63-wide band (K-window of 80 channels per 16-channel d-tile).
//
// Round-2 changes:
//  * LDS tile stores 576 rows (channels unwrapped: row r holds channel
//    (r-32) mod 512) so B addressing is affine -> all 20 ds_load pairs get
//    immediate offsets and batch ahead of the WMMA chain (no per-WMMA
//    s_wait_dscnt 0 stalls).
//  * Tile fill uses CDNA5 async copies: global_load_async_to_lds_b128
//    (ASYNCcnt) drained with s_wait_asynccnt before the barrier.

typedef __attribute__((ext_vector_type(2))) float v2f;
typedef __attribute__((ext_vector_type(4))) float v4f;
typedef __attribute__((ext_vector_type(8))) float v8f;

#define C_CH   512
#define HW     3136          // 56*56
#define NT     64            // spatial columns per workgroup tile
#define ROWS   576           // 512 channels + 2*32 halo (circulant unwrap)
#define KSTEPS 20            // 80-channel band window / K=4
#define FILT_TAPS 63
#define USE_ASYNC_LDS 1

extern "C" __global__ __launch_bounds__(256)
void toeplitz_wmma_f32_kernel(const float* __restrict__ act,
                              const float* __restrict__ filt,
                              float* __restrict__ out)
{
    extern __shared__ float lds[];           // [ROWS][NT] tile + filter taps
    float* tile = lds;                       // ROWS*NT floats = 144 KB
    float* fsh  = lds + ROWS * NT;           // 63 floats (+pad)

    const int tid  = threadIdx.x;            // 0..255 (8 waves of 32)
    const int lane = tid & 31;
    const int wave = tid >> 5;
    const int b    = blockIdx.y;             // batch image
    const int n0   = blockIdx.x * NT;        // spatial column block

    // ---- stage the 63-tap Ricker filter into LDS ----
    if (tid < FILT_TAPS) fsh[tid] = filt[tid];

    // ---- tile fill: row r <- channel (r - 32) mod 512, columns [n0, n0+64) ----
    // 16 lanes * 16B cover one row; 256 threads -> 16 rows per pass, 36 passes.
    {
        const int rowInPass = tid >> 4;          // 0..15
        const int colv      = (tid & 15) << 2;   // 0,4,...,60
        const float* gb     = act + (size_t)b * C_CH * HW;   // uniform SGPR base
        const unsigned coloff = (unsigned)(n0 + colv) * 4u;
#if USE_ASYNC_LDS
        for (int pass = 0; pass < ROWS / 16; ++pass) {
            const int r  = (pass << 4) + rowInPass;
            const int ch = (r + C_CH - 32) & (C_CH - 1);
            const unsigned goff  = (unsigned)(ch * HW) * 4u + coloff;  // byte offset
            const unsigned laddr = (unsigned)(size_t)(tile + r * NT + colv); // LDS addr
            asm volatile("global_load_async_to_lds_b128 %0, %1, %2"
                         :: "v"(laddr), "v"(goff), "s"(gb) : "memory");
        }
        asm volatile("s_wait_asynccnt 0x0" ::: "memory");
#else
        for (int pass = 0; pass < ROWS / 16; ++pass) {
            const int r  = (pass << 4) + rowInPass;
            const int ch = (r + C_CH - 32) & (C_CH - 1);
            v4f v = *(const v4f*)(gb + (size_t)ch * HW + n0 + colv);
            *(v4f*)(tile + r * NT + colv) = v;
        }
#endif
    }
    __syncthreads();

    // ---- precompute Toeplitz A-fragments: identical for ALL d-tiles ----
    // A[m,kk] = (0 <= 63+m-kk <= 62) ? filt[63+m-kk] : 0, kk in [0,80)
    const int mOrN  = lane & 15;                 // M for A, N for B
    const int khalf = (lane >> 4) << 1;          // 0 or 2 (K half per lane group)
    v2f afrag[KSTEPS];
    #pragma unroll
    for (int s = 0; s < KSTEPS; ++s) {
        const int kk0 = 4 * s + khalf;
        const int i0  = 63 + mOrN - kk0;         // tap index for a.x
        const int i1  = i0 - 1;                  // tap index for a.y (kk0+1)
        afrag[s].x = (i0 >= 0 && i0 < FILT_TAPS) ? fsh[i0] : 0.0f;
        afrag[s].y = (i1 >= 0 && i1 < FILT_TAPS) ? fsh[i1] : 0.0f;
    }

    // ---- each wave computes 16 of the 128 (d-tile, n-tile) 16x16 outputs ----
    float* outb = out + (size_t)b * C_CH * HW + n0;
    for (int i = 0; i < 16; ++i) {
        const int t     = wave + (i << 3);       // 0..127
        const int dtile = t >> 2;                // 0..31
        const int ntile = t & 3;                 // 0..3
        const int d0    = dtile << 4;
        const int ncol  = (ntile << 4) + mOrN;   // this lane's B column in LDS

        // B rows are now affine: LDS row d0+kk holds channel d0+kk-32 (mod 512)
        v8f acc = {};
        #pragma unroll
        for (int s = 0; s < KSTEPS; ++s) {
            const int kk0 = 4 * s + khalf;
            v2f bfrag;
            bfrag.x = tile[(d0 + kk0)     * NT + ncol];
            bfrag.y = tile[(d0 + kk0 + 1) * NT + ncol];
            acc = __builtin_amdgcn_wmma_f32_16x16x4_f32(
                      /*neg_a=*/false, afrag[s],
                      /*neg_b=*/false, bfrag,
                      /*c_mod=*/(short)0, acc,
                      /*reuse_a=*/false, /*reuse_b=*/false);
        }

        // D layout: VGPR r -> row d0 + r + (lane<16 ? 0 : 8), col = lane%16
        const int mrow = d0 + ((lane >> 4) << 3);
        float* op = outb + (size_t)mrow * HW + (ntile << 4) + mOrN;
        #pragma unroll
        for (int r = 0; r < 8; ++r)
            op[(size_t)r * HW] = acc[r];
    }
}

extern "C" void kernel_launch(void* const* d_in, const int* in_sizes, int n_in,
                              void* d_out, int out_size, void* d_ws, size_t ws_size,
                              hipStream_t stream) {
    const float* act  = (const float*)d_in[0];   // (32, 512, 56, 56) f32
    const float* filt = (const float*)d_in[1];   // (1, 1, 63) f32
    float*       outp = (float*)d_out;           // (32, 512, 56, 56) f32

    const int batches = in_sizes[0] / (C_CH * HW);   // 32
    dim3 grid(HW / NT, batches, 1);                  // 49 x 32 workgroups
    dim3 block(256, 1, 1);                           // 8 waves (wave32)
    size_t shmem = (size_t)(ROWS * NT + 64) * sizeof(float);  // 144 KB + filter

    toeplitz_wmma_f32_kernel<<<grid, block, shmem, stream>>>(act, filt, outp);
}